// SmoothAttnDecoder_1889785610894
// MI455X (gfx1250) — compile-verified
//
#include <hip/hip_runtime.h>
#include <hip/hip_bf16.h>

#define LANG   50257
#define EDIM   512
#define HDIM   1024
#define ADIM   512
#define SLEN   1024
#define BOUND  25
#define TEMP_F 10.0f
#define SOS_I  1

typedef float v2f __attribute__((ext_vector_type(2)));
typedef float v8f __attribute__((ext_vector_type(8)));

#if defined(__has_builtin)
#if __has_builtin(__builtin_amdgcn_wmma_f32_16x16x4_f32)
#define HAVE_WMMA_F32X4 1
#endif
#endif

__device__ __forceinline__ float waveReduceSum(float v) {
  #pragma unroll
  for (int off = 16; off >= 1; off >>= 1) v += __shfl_xor(v, off, 32);
  return v;
}
__device__ __forceinline__ int imin(int a, int b) { return a < b ? a : b; }
__device__ __forceinline__ float dot4(float4 a, float4 b) {
  return a.x * b.x + a.y * b.y + a.z * b.z + a.w * b.w;
}

// ---------------------------------------------------------------------------
// init: h0 = hidden[-1]; emb0 = emb_T[SOS] (word0 is one-hot at SOS)
// ---------------------------------------------------------------------------
__global__ void k_init(const float* __restrict__ hidden, const float* __restrict__ T,
                       float* __restrict__ h0, float* __restrict__ emb0) {
  int t = blockIdx.x * 256 + threadIdx.x;  // grid 4*256 = 1024
  if (t < HDIM) h0[t] = hidden[t];
  if (t < EDIM) emb0[t] = T[(size_t)SOS_I * EDIM + t];
}

// ---------------------------------------------------------------------------
// Precompute P[s][a] = embeddings(1024x512) @ attn_W1[0:512,:] + b1   (WMMA)
// One wave per 16x16 tile; V_WMMA_F32_16X16X4_F32 over K=512.
// ---------------------------------------------------------------------------
__global__ void k_precompute_P(const float* __restrict__ emb,  // (1024,512)
                               const float* __restrict__ W1,   // (1536,512)
                               const float* __restrict__ b1,   // (512)
                               float* __restrict__ P) {        // (1024,512)
  const int lane  = threadIdx.x & 31;
  const int wave  = threadIdx.x >> 5;
  const int tile  = blockIdx.x * 8 + wave;   // 2048 tiles total
  const int sTile = tile >> 5;               // 64 tiles along S
  const int aTile = tile & 31;               // 32 tiles along A
  const int mrow  = sTile * 16 + (lane & 15);
  const int ncol  = aTile * 16 + (lane & 15);
  const int hi    = lane >> 4;               // half-wave select
  v8f c = {};
#if HAVE_WMMA_F32X4
  for (int k = 0; k < EDIM; k += 4) {
    const int kk = k + 2 * hi;
    v2f a, b;
    // A (16x4 f32): lanes 0-15 -> K=k,k+1 ; lanes 16-31 -> K=k+2,k+3 ; M=lane%16
    a = *(const v2f*)(emb + (size_t)mrow * EDIM + kk);
    // B (4x16 f32): same K split, N=lane%16
    b.x = W1[(size_t)kk * ADIM + ncol];
    b.y = W1[(size_t)(kk + 1) * ADIM + ncol];
    c = __builtin_amdgcn_wmma_f32_16x16x4_f32(false, a, false, b,
                                              (short)0, c, false, false);
  }
#else
  // VALU fallback producing the same D layout
  #pragma unroll
  for (int i = 0; i < 8; ++i) {
    const int m = sTile * 16 + i + hi * 8;
    float acc = 0.f;
    for (int k = 0; k < EDIM; ++k)
      acc += emb[(size_t)m * EDIM + k] * W1[(size_t)k * ADIM + ncol];
    c[i] = acc;
  }
#endif
  const float bias = b1[ncol];
  // D layout: VGPR i -> (M = i + 8*hi, N = lane%16)
  #pragma unroll
  for (int i = 0; i < 8; ++i) {
    const int m = sTile * 16 + i + hi * 8;
    P[(size_t)m * ADIM + ncol] = c[i] + bias;
  }
}

// ---------------------------------------------------------------------------
// GRU gates: gi = W_ih@emb + b_ih (3072x512); gh = W_hh@h + b_hh (3072x1024)
// One wave per output row; float4 lane-strided coalesced reads (b128).
// ---------------------------------------------------------------------------
__global__ void k_gates(const float* __restrict__ emb, const float* __restrict__ h,
                        const float* __restrict__ W_ih, const float* __restrict__ W_hh,
                        const float* __restrict__ b_ih, const float* __restrict__ b_hh,
                        float* __restrict__ gi, float* __restrict__ gh) {
  const int lane = threadIdx.x & 31;
  const int r    = blockIdx.x * 8 + (threadIdx.x >> 5);  // 0..6143
  float acc = 0.f;
  if (r < 3 * HDIM) {
    const float4* wr = (const float4*)(W_ih + (size_t)r * EDIM);
    const float4* xv = (const float4*)emb;
    #pragma unroll
    for (int j = lane; j < EDIM / 4; j += 32) acc += dot4(wr[j], xv[j]);
    acc = waveReduceSum(acc);
    if (lane == 0) gi[r] = acc + b_ih[r];
  } else {
    const int rr = r - 3 * HDIM;
    const float4* wr = (const float4*)(W_hh + (size_t)rr * HDIM);
    const float4* xv = (const float4*)h;
    #pragma unroll
    for (int j = lane; j < HDIM / 4; j += 32) acc += dot4(wr[j], xv[j]);
    acc = waveReduceSum(acc);
    if (lane == 0) gh[rr] = acc + b_hh[rr];
  }
}

__global__ void k_gru_combine(const float* __restrict__ gi, const float* __restrict__ gh,
                              const float* __restrict__ hprev, float* __restrict__ hnew) {
  const int i = blockIdx.x * 256 + threadIdx.x;  // < 1024
  const float r = 1.f / (1.f + __expf(-(gi[i] + gh[i])));
  const float z = 1.f / (1.f + __expf(-(gi[HDIM + i] + gh[HDIM + i])));
  const float n = tanhf(gi[2 * HDIM + i] + r * gh[2 * HDIM + i]);
  hnew[i] = (1.f - z) * n + z * hprev[i];
}

// qproj[a] = sum_k h[k] * W1[(512+k)*512 + a]   (column access, coalesced over a)
__global__ void k_qproj(const float* __restrict__ h, const float* __restrict__ W1,
                        float* __restrict__ qproj) {
  const int a = blockIdx.x * 256 + threadIdx.x;  // < 512
  float acc = 0.f;
  for (int k = 0; k < HDIM; ++k)
    acc += h[k] * W1[(size_t)(ADIM + k) * ADIM + a];
  qproj[a] = acc;
}

// scores[s] = sum_a tanh(P[s,a] + qproj[a]) * v[a];  one wave per s, b128 loads
__global__ void k_scores(const float* __restrict__ P, const float* __restrict__ qproj,
                         const float* __restrict__ vvec, float* __restrict__ scores) {
  const int lane = threadIdx.x & 31;
  const int s    = blockIdx.x * 8 + (threadIdx.x >> 5);  // < 1024
  const float4* pr = (const float4*)(P + (size_t)s * ADIM);
  const float4* qp = (const float4*)qproj;
  const float4* vv = (const float4*)vvec;
  float acc = 0.f;
  #pragma unroll
  for (int a = lane; a < ADIM / 4; a += 32) {
    float4 p = pr[a], q = qp[a], v = vv[a];
    acc += tanhf(p.x + q.x) * v.x + tanhf(p.y + q.y) * v.y +
           tanhf(p.z + q.z) * v.z + tanhf(p.w + q.w) * v.w;
  }
  acc = waveReduceSum(acc);
  if (lane == 0) scores[s] = acc;
}

// softmax over 1024 scores + context vector, single block of 512 threads
__global__ void k_attn(const float* __restrict__ scores, const float* __restrict__ embeddings,
                       float* __restrict__ attn, float* __restrict__ wout) {
  __shared__ float w[SLEN];
  __shared__ float red[512];
  const int t = threadIdx.x;
  float m = fmaxf(scores[t], scores[t + 512]);
  red[t] = m; __syncthreads();
  for (int s = 256; s; s >>= 1) { if (t < s) red[t] = fmaxf(red[t], red[t + s]); __syncthreads(); }
  m = red[0]; __syncthreads();
  const float e0 = __expf(scores[t] - m), e1 = __expf(scores[t + 512] - m);
  red[t] = e0 + e1; __syncthreads();
  for (int s = 256; s; s >>= 1) { if (t < s) red[t] += red[t + s]; __syncthreads(); }
  const float inv = 1.f / red[0];
  w[t] = e0 * inv; w[t + 512] = e1 * inv;
  __syncthreads();
  wout[t] = w[t]; wout[t + 512] = w[t + 512];
  float acc = 0.f;
  for (int s = 0; s < SLEN; ++s)
    acc += w[s] * embeddings[(size_t)s * ADIM + t];
  attn[t] = acc;  // t < 512
}

// vect[l] = TEMP * (out_b[l] + out_W[l,:] . [h ; attn]); one wave per row.
// float4 lane accesses -> global_load_b128, 512B contiguous per wave per issue.
__global__ void k_outproj(const float* __restrict__ h, const float* __restrict__ attn,
                          const float* __restrict__ out_W, const float* __restrict__ out_b,
                          float* __restrict__ vect) {
  __shared__ __align__(16) float merge[HDIM + ADIM];
  for (int i = threadIdx.x; i < HDIM + ADIM; i += 256)
    merge[i] = (i < HDIM) ? h[i] : attn[i - HDIM];
  __syncthreads();
  const int lane = threadIdx.x & 31;
  const int l    = blockIdx.x * 8 + (threadIdx.x >> 5);
  if (l >= LANG) return;  // wave-uniform exit
  const float4* wr = (const float4*)(out_W + (size_t)l * (HDIM + ADIM));
  const float4* mg = (const float4*)merge;
  float acc = 0.f;
  #pragma unroll
  for (int j = lane; j < (HDIM + ADIM) / 4; j += 32)  // 12 iterations
    acc += dot4(wr[j], mg[j]);
  acc = waveReduceSum(acc);
  if (lane == 0) vect[l] = (acc + out_b[l]) * TEMP_F;
}

// vocab softmax, pass 1: per-chunk (max, sum-exp) partials.  64 chunks of 786.
__global__ void k_softmax_part(const float* __restrict__ vect,
                               float* __restrict__ pmax, float* __restrict__ psum) {
  __shared__ float red[256];
  const int start = blockIdx.x * 786;
  const int end   = imin(start + 786, LANG);
  float m = -1e30f;
  for (int i = start + threadIdx.x; i < end; i += 256) m = fmaxf(m, vect[i]);
  red[threadIdx.x] = m; __syncthreads();
  for (int s = 128; s; s >>= 1) { if (threadIdx.x < s) red[threadIdx.x] = fmaxf(red[threadIdx.x], red[threadIdx.x + s]); __syncthreads(); }
  m = red[0]; __syncthreads();
  float sum = 0.f;
  for (int i = start + threadIdx.x; i < end; i += 256) sum += __expf(vect[i] - m);
  red[threadIdx.x] = sum; __syncthreads();
  for (int s = 128; s; s >>= 1) { if (threadIdx.x < s) red[threadIdx.x] += red[threadIdx.x + s]; __syncthreads(); }
  if (threadIdx.x == 0) { pmax[blockIdx.x] = m; psum[blockIdx.x] = red[0]; }
}

// pass 2: merge 64 partials into global (M, Z)
__global__ void k_softmax_merge(const float* __restrict__ pmax, const float* __restrict__ psum,
                                float* __restrict__ MZ) {
  __shared__ float rm[64], rs[64];
  const int t = threadIdx.x;  // 64 threads
  rm[t] = pmax[t]; __syncthreads();
  for (int s = 32; s; s >>= 1) { if (t < s) rm[t] = fmaxf(rm[t], rm[t + s]); __syncthreads(); }
  const float M = rm[0];
  rs[t] = psum[t] * __expf(pmax[t] - M); __syncthreads();
  for (int s = 32; s; s >>= 1) { if (t < s) rs[t] += rs[t + s]; __syncthreads(); }
  if (t == 0) { MZ[0] = M; MZ[1] = rs[0]; }
}

// pass 3: word[v] = exp(vect[v] - M) / Z
__global__ void k_word(const float* __restrict__ vect, const float* __restrict__ MZ,
                       float* __restrict__ word) {
  const int i = blockIdx.x * 256 + threadIdx.x;
  if (i < LANG) word[i] = __expf(vect[i] - MZ[0]) / MZ[1];
}

// emb GEMV pass 1: partial[c][e] = sum_{v in chunk c} word[v] * T[v][e]
// 128 chunks of 393 rows; thread handles a float2 column pair (b64 loads).
__global__ void k_emb_part(const float* __restrict__ word, const float* __restrict__ T,
                           float* __restrict__ partial) {
  const int c  = blockIdx.x;
  const int v0 = c * 393;
  const int v1 = imin(v0 + 393, LANG);
  const int e  = threadIdx.x;  // float2 index, 256 covers 512 floats
  float2 acc = make_float2(0.f, 0.f);
  for (int v = v0; v < v1; ++v) {
    const float wv = word[v];
    const float2 r = ((const float2*)(T + (size_t)v * EDIM))[e];
    acc.x += wv * r.x;
    acc.y += wv * r.y;
  }
  ((float2*)(partial + (size_t)c * EDIM))[e] = acc;
}

// emb GEMV pass 2: deterministic reduce over 128 chunks (float2 lanes)
__global__ void k_emb_reduce(const float* __restrict__ partial, float* __restrict__ emb) {
  const int e = threadIdx.x;  // single block of 256, float2 index
  float2 acc = make_float2(0.f, 0.f);
  for (int c = 0; c < 128; ++c) {
    const float2 p = ((const float2*)(partial + (size_t)c * EDIM))[e];
    acc.x += p.x;
    acc.y += p.y;
  }
  ((float2*)emb)[e] = acc;
}

// ---------------------------------------------------------------------------
extern "C" void kernel_launch(void* const* d_in, const int* in_sizes, int n_in,
                              void* d_out, int out_size, void* d_ws, size_t ws_size,
                              hipStream_t stream) {
  const float* hidden     = (const float*)d_in[0];
  const float* embeddings = (const float*)d_in[1];
  const float* emb_T      = (const float*)d_in[2];
  const float* W_ih       = (const float*)d_in[3];
  const float* W_hh       = (const float*)d_in[4];
  const float* b_ih       = (const float*)d_in[5];
  const float* b_hh       = (const float*)d_in[6];
  const float* attn_W1    = (const float*)d_in[7];
  const float* attn_b1    = (const float*)d_in[8];
  const float* attn_v     = (const float*)d_in[9];
  const float* out_W      = (const float*)d_in[10];
  const float* out_b      = (const float*)d_in[11];

  float* outp    = (float*)d_out;
  float* vects   = outp;                               // (25, 50257)
  float* weights = outp + (size_t)BOUND * LANG;        // (1, 25, 1024)

  float* ws      = (float*)d_ws;
  float* P       = ws;  ws += (size_t)SLEN * ADIM;     // 524288
  float* hbuf    = ws;  ws += 2 * HDIM;                // ping-pong h
  float* emb     = ws;  ws += EDIM;
  float* gi      = ws;  ws += 3 * HDIM;
  float* gh      = ws;  ws += 3 * HDIM;
  float* qproj   = ws;  ws += ADIM;
  float* scores  = ws;  ws += SLEN;
  float* attn    = ws;  ws += ADIM;
  float* pmax    = ws;  ws += 64;
  float* psum    = ws;  ws += 64;
  float* MZ      = ws;  ws += 2;
  float* word    = ws;  ws += LANG;
  float* partial = ws;  ws += 128 * EDIM;

  k_init<<<4, 256, 0, stream>>>(hidden, emb_T, hbuf, emb);
  k_precompute_P<<<256, 256, 0, stream>>>(embeddings, attn_W1, attn_b1, P);

  for (int t = 0; t < BOUND; ++t) {
    float* hprev = hbuf + (t & 1) * HDIM;
    float* hnew  = hbuf + ((t + 1) & 1) * HDIM;
    float* vect  = vects + (size_t)t * LANG;

    k_gates<<<768, 256, 0, stream>>>(emb, hprev, W_ih, W_hh, b_ih, b_hh, gi, gh);
    k_gru_combine<<<4, 256, 0, stream>>>(gi, gh, hprev, hnew);
    k_qproj<<<2, 256, 0, stream>>>(hnew, attn_W1, qproj);
    k_scores<<<128, 256, 0, stream>>>(P, qproj, attn_v, scores);
    k_attn<<<1, 512, 0, stream>>>(scores, embeddings, attn, weights + (size_t)t * SLEN);
    k_outproj<<<(LANG + 7) / 8, 256, 0, stream>>>(hnew, attn, out_W, out_b, vect);

    if (t < BOUND - 1) {
      k_softmax_part<<<64, 256, 0, stream>>>(vect, pmax, psum);
      k_softmax_merge<<<1, 64, 0, stream>>>(pmax, psum, MZ);
      k_word<<<(LANG + 255) / 256, 256, 0, stream>>>(vect, MZ, word);
      k_emb_part<<<128, 256, 0, stream>>>(word, emb_T, partial);
      k_emb_reduce<<<1, 256, 0, stream>>>(partial, emb);
    }
  }
}